// CycleFC_1382979469471
// MI455X (gfx1250) — compile-verified
//
#include <hip/hip_runtime.h>
#include <stdint.h>

// CycleFC channel-cyclic shift along W, collapsed to:
//   out[b,c,h,w] = (0 <= w-s(c) < 64) ? x[b,c,h, w-s(c)] : 0
//   s(c) = 2 - |(c&7) - 4|   (shift list [-2,-1,0,1,2,1,0,-1] cycled over C=256)
//
// Pure bandwidth kernel: 256 MiB traffic, ~11.5 us at 23.3 TB/s HBM. Strategy:
//  - 1 block per (b,c): a contiguous 16 KiB slab of 64 rows; shift is block-uniform (SALU only).
//  - Stage slab into LDS with CDNA5 async copies (GLOBAL_LOAD_ASYNC_TO_LDS_B128 / ASYNCcnt).
//  - 2-float ZERO HALO per LDS row makes the shifted read branch-free (no exec-mask
//    divergence, no cndmask): every index w-s lands on data or pre-zeroed halo.
//  - Write out with aligned nontemporal global_store_b128 (in+out streams > 192MB L2,
//    zero reuse -> keep them out of the way).

typedef float v4f __attribute__((ext_vector_type(4)));   // NT-store-compatible float4

#define TILE_W      64
#define TILE_ROWS   64
#define THREADS     256
#define TILE_ELEMS  (TILE_ROWS * TILE_W)   // 4096 floats = 16 KiB per slab
#define LDS_STRIDE  72                     // floats per LDS row (288 B, 16B-multiple)
#define INTERIOR    4                      // interior starts at float 4 (byte 16) of each row

__global__ __launch_bounds__(THREADS) void cyclefc_shift_kernel(
    const float* __restrict__ x, float* __restrict__ out)
{
    __shared__ __align__(16) float tile[TILE_ROWS * LDS_STRIDE];  // 18 KiB

    const int bc  = blockIdx.x;            // bc = b*256 + c (one (b,c) slab per block)
    const int c   = bc & 255;
    const int s   = 2 - abs((c & 7) - 4);  // block-uniform shift along W, |s| <= 2
    const int tid = threadIdx.x;

    const float* src = x   + (size_t)bc * TILE_ELEMS;
    float*       dst = out + (size_t)bc * TILE_ELEMS;

    // ---- Zero halo: floats {2,3,68,69} of each row; 64 rows * 4 = 256 = one store/thread
    {
        const int hrow = tid >> 2;
        const int p    = tid & 3;                       // 0,1 -> left halo; 2,3 -> right halo
        const int hoff = 2 + (p & 1) + ((p >> 1) * 66); // {2,3,68,69}
        tile[hrow * LDS_STRIDE + hoff] = 0.0f;
    }

    // ---- Stage 1: async global -> LDS copy of the 16 KiB slab -------------
    // Each lane moves 16 B per issue into the 16B-aligned row interior.
    // Global and LDS row strides differ (256 B vs 288 B), so each issue gets
    // its own address pair (INST_OFFSET applies to both addresses equally).
    {
        const uint32_t lds_base = (uint32_t)(uintptr_t)(&tile[0]);
#pragma unroll
        for (int k = 0; k < 4; ++k) {
            const int t = k * THREADS + tid;                     // b128 chunk id 0..1023
            const uint32_t g = (uint32_t)t * 16u;                // global byte offset
            const uint32_t l = lds_base
                             + (uint32_t)(t >> 4) * (LDS_STRIDE * 4)  // row * 288
                             + (uint32_t)(INTERIOR * 4)              // +16
                             + (uint32_t)(t & 15) * 16u;
            asm volatile("global_load_async_to_lds_b128 %0, %1, %2"
                         :
                         : "v"(l), "v"(g), "s"(src)
                         : "memory");
        }
        asm volatile("s_wait_asynccnt 0x0" ::: "memory");
    }
    __syncthreads();

    // ---- Stage 2: branch-free shifted read from LDS, NT b128 store --------
    // 1024 float4 output chunks per slab; 4 per thread. Index range of
    // (INTERIOR + w - s) is [2, 69] -> always inside data or zero halo.
#pragma unroll
    for (int k = 0; k < 4; ++k) {
        const int t   = k * THREADS + tid;   // chunk id 0..1023
        const int row = t >> 4;              // 0..63
        const int w0  = (t & 15) << 2;       // 0,4,...,60
        const float* p = &tile[row * LDS_STRIDE + INTERIOR + w0 - s];

        v4f v;
        v.x = p[0];
        v.y = p[1];
        v.z = p[2];
        v.w = p[3];

        __builtin_nontemporal_store(v, reinterpret_cast<v4f*>(dst + (size_t)t * 4));
    }
}

extern "C" void kernel_launch(void* const* d_in, const int* in_sizes, int n_in,
                              void* d_out, int out_size, void* d_ws, size_t ws_size,
                              hipStream_t stream)
{
    (void)n_in; (void)out_size; (void)d_ws; (void)ws_size;
    const float* x   = (const float*)d_in[0];
    float*       out = (float*)d_out;

    // Total elements / (H*W = 4096) = number of (b,c) slabs = 32*256 = 8192.
    const int slabs = in_sizes[0] / TILE_ELEMS;

    cyclefc_shift_kernel<<<slabs, THREADS, 0, stream>>>(x, out);
}